// TransD_71519795413416
// MI455X (gfx1250) — compile-verified
//
#include <hip/hip_runtime.h>
#include <hip/hip_bf16.h>

typedef __attribute__((ext_vector_type(2))) float v2f;
typedef __attribute__((ext_vector_type(8))) float v8f;

#define D_DIM           256
#define ELEMS_PER_LANE  8      // 256 / 32 lanes
#define BLOCK_THREADS   256
#define WAVES_PER_BLOCK 8      // wave32
#define NBLOCKS         1024

// Full wave32 sum via V_WMMA_F32_16X16X4_F32.
// A (16x4 f32, 2 VGPRs): VGPR0 = v, VGPR1 = 0
//   => A[m][0] = v[lane m], A[m][2] = v[lane m+16], A[m][1] = A[m][3] = 0
// B = all-ones: invariant under B's VGPR layout => D[m][n] = v[m] + v[m+16].
// C/D layout: lane n (<16) holds D[0..7][n] in its 8 regs, lane n+16 holds
// D[8..15][n]. So sum-of-8-regs + one shfl_xor(16) = exact f32 wave total.
// REQUIRES EXEC all-1s at the call site.
__device__ __forceinline__ float wave_reduce_wmma(float v) {
    v2f a;    a.x = v;      a.y = 0.0f;
    v2f ones; ones.x = 1.0f; ones.y = 1.0f;
    v8f c = {};
    v8f d = __builtin_amdgcn_wmma_f32_16x16x4_f32(
        /*neg_a=*/false, a, /*neg_b=*/false, ones,
        /*c_mod=*/(short)0, c, /*reuse_a=*/false, /*reuse_b=*/false);
    float s01 = d[0] + d[1];
    float s23 = d[2] + d[3];
    float s45 = d[4] + d[5];
    float s67 = d[6] + d[7];
    float s = (s01 + s23) + (s45 + s67);
    s += __shfl_xor(s, 16, 32);
    return s;
}

__global__ __launch_bounds__(BLOCK_THREADS)
void transd_partial_kernel(const int* __restrict__ head_ids,
                           const int* __restrict__ meta_head_ids,
                           const int* __restrict__ relation_ids,
                           const int* __restrict__ meta_relation_ids,
                           const int* __restrict__ tail_ids,
                           const int* __restrict__ meta_tail_ids,
                           const float* __restrict__ entity_emb,
                           const float* __restrict__ relation_emb,
                           const float* __restrict__ meta_entity_emb,
                           const float* __restrict__ meta_relation_emb,
                           float* __restrict__ block_partials,
                           int B)
{
    const int lane  = threadIdx.x & 31;
    const int wave  = threadIdx.x >> 5;
    const int gwave = blockIdx.x * WAVES_PER_BLOCK + wave;
    const int nwave = gridDim.x * WAVES_PER_BLOCK;
    const int e0    = lane * ELEMS_PER_LANE;

    float acc = 0.0f;   // per-lane partial of sum(|diff|)

    // `row` is identical across the wave -> uniform branch, EXEC stays all-1s
    // inside the loop body (required for the WMMA reductions).
    for (int row = gwave; row < B; row += nwave) {
        const float* hrow  = entity_emb        + (size_t)head_ids[row]          * D_DIM;
        const float* mhrow = meta_entity_emb   + (size_t)meta_head_ids[row]     * D_DIM;
        const float* rrow  = relation_emb      + (size_t)relation_ids[row]      * D_DIM;
        const float* mrrow = meta_relation_emb + (size_t)meta_relation_ids[row] * D_DIM;
        const float* trow  = entity_emb        + (size_t)tail_ids[row]          * D_DIM;
        const float* mtrow = meta_entity_emb   + (size_t)meta_tail_ids[row]     * D_DIM;

        // each lane: 8 contiguous floats (two float4 loads) of each of 6 rows
        float4 ha  = *(const float4*)(hrow  + e0);  float4 hb  = *(const float4*)(hrow  + e0 + 4);
        float4 mha = *(const float4*)(mhrow + e0);  float4 mhb = *(const float4*)(mhrow + e0 + 4);
        float4 ta  = *(const float4*)(trow  + e0);  float4 tb  = *(const float4*)(trow  + e0 + 4);
        float4 mta = *(const float4*)(mtrow + e0);  float4 mtb = *(const float4*)(mtrow + e0 + 4);
        float4 ra  = *(const float4*)(rrow  + e0);  float4 rb  = *(const float4*)(rrow  + e0 + 4);
        float4 mra = *(const float4*)(mrrow + e0);  float4 mrb = *(const float4*)(mrrow + e0 + 4);

        // per-lane dot partials
        float d1p = 0.0f, d2p = 0.0f;
        d1p = fmaf(ha.x, mha.x, d1p); d1p = fmaf(ha.y, mha.y, d1p);
        d1p = fmaf(ha.z, mha.z, d1p); d1p = fmaf(ha.w, mha.w, d1p);
        d1p = fmaf(hb.x, mhb.x, d1p); d1p = fmaf(hb.y, mhb.y, d1p);
        d1p = fmaf(hb.z, mhb.z, d1p); d1p = fmaf(hb.w, mhb.w, d1p);
        d2p = fmaf(ta.x, mta.x, d2p); d2p = fmaf(ta.y, mta.y, d2p);
        d2p = fmaf(ta.z, mta.z, d2p); d2p = fmaf(ta.w, mta.w, d2p);
        d2p = fmaf(tb.x, mtb.x, d2p); d2p = fmaf(tb.y, mtb.y, d2p);
        d2p = fmaf(tb.w, mtb.w, fmaf(tb.z, mtb.z, d2p));

        float d1 = wave_reduce_wmma(d1p);   // dot(mh, h)
        float d2 = wave_reduce_wmma(d2p);   // dot(mt, t)

        // rank-1 projection: p = mr * dot + x
        float4 pha, phb, pta, ptb;
        pha.x = fmaf(mra.x, d1, ha.x); pha.y = fmaf(mra.y, d1, ha.y);
        pha.z = fmaf(mra.z, d1, ha.z); pha.w = fmaf(mra.w, d1, ha.w);
        phb.x = fmaf(mrb.x, d1, hb.x); phb.y = fmaf(mrb.y, d1, hb.y);
        phb.z = fmaf(mrb.z, d1, hb.z); phb.w = fmaf(mrb.w, d1, hb.w);
        pta.x = fmaf(mra.x, d2, ta.x); pta.y = fmaf(mra.y, d2, ta.y);
        pta.z = fmaf(mra.z, d2, ta.z); pta.w = fmaf(mra.w, d2, ta.w);
        ptb.x = fmaf(mrb.x, d2, tb.x); ptb.y = fmaf(mrb.y, d2, tb.y);
        ptb.z = fmaf(mrb.z, d2, tb.z); ptb.w = fmaf(mrb.w, d2, tb.w);

        // squared-norm partials
        float nhp = 0.0f, ntp = 0.0f;
        nhp = fmaf(pha.x, pha.x, nhp); nhp = fmaf(pha.y, pha.y, nhp);
        nhp = fmaf(pha.z, pha.z, nhp); nhp = fmaf(pha.w, pha.w, nhp);
        nhp = fmaf(phb.x, phb.x, nhp); nhp = fmaf(phb.y, phb.y, nhp);
        nhp = fmaf(phb.z, phb.z, nhp); nhp = fmaf(phb.w, phb.w, nhp);
        ntp = fmaf(pta.x, pta.x, ntp); ntp = fmaf(pta.y, pta.y, ntp);
        ntp = fmaf(pta.z, pta.z, ntp); ntp = fmaf(pta.w, pta.w, ntp);
        ntp = fmaf(ptb.x, ptb.x, ntp); ntp = fmaf(ptb.y, ptb.y, ntp);
        ntp = fmaf(ptb.w, ptb.w, fmaf(ptb.z, ptb.z, ntp));

        float nh2 = wave_reduce_wmma(nhp);
        float nt2 = wave_reduce_wmma(ntp);

        // F.normalize: x / max(||x||, 1e-12)
        float rnh = 1.0f / fmaxf(sqrtf(nh2), 1e-12f);
        float rnt = 1.0f / fmaxf(sqrtf(nt2), 1e-12f);

        // |proj_h/nh + r - proj_t/nt|
        acc += fabsf(fmaf(pha.x, rnh, ra.x) - pta.x * rnt);
        acc += fabsf(fmaf(pha.y, rnh, ra.y) - pta.y * rnt);
        acc += fabsf(fmaf(pha.z, rnh, ra.z) - pta.z * rnt);
        acc += fabsf(fmaf(pha.w, rnh, ra.w) - pta.w * rnt);
        acc += fabsf(fmaf(phb.x, rnh, rb.x) - ptb.x * rnt);
        acc += fabsf(fmaf(phb.y, rnh, rb.y) - ptb.y * rnt);
        acc += fabsf(fmaf(phb.z, rnh, rb.z) - ptb.z * rnt);
        acc += fabsf(fmaf(phb.w, rnh, rb.w) - ptb.w * rnt);
    }

    // wave total of |diff| partials (EXEC all-1s again after uniform loop exit)
    float s = wave_reduce_wmma(acc);

    __shared__ float wsum[WAVES_PER_BLOCK];
    if (lane == 0) wsum[wave] = s;
    __syncthreads();
    if (threadIdx.x == 0) {
        float bs = 0.0f;
        #pragma unroll
        for (int i = 0; i < WAVES_PER_BLOCK; ++i) bs += wsum[i];
        block_partials[blockIdx.x] = bs;
    }
}

__global__ __launch_bounds__(BLOCK_THREADS)
void final_reduce_kernel(const float* __restrict__ block_partials,
                         float* __restrict__ out, int n)
{
    __shared__ float sm[BLOCK_THREADS];
    float s = 0.0f;
    for (int i = threadIdx.x; i < n; i += BLOCK_THREADS) s += block_partials[i];
    sm[threadIdx.x] = s;
    __syncthreads();
    for (int off = BLOCK_THREADS / 2; off > 0; off >>= 1) {
        if (threadIdx.x < off) sm[threadIdx.x] += sm[threadIdx.x + off];
        __syncthreads();
    }
    if (threadIdx.x == 0) out[0] = sm[0];
}

extern "C" void kernel_launch(void* const* d_in, const int* in_sizes, int n_in,
                              void* d_out, int out_size, void* d_ws, size_t ws_size,
                              hipStream_t stream) {
    const int*   head_ids          = (const int*)d_in[0];
    const int*   meta_head_ids     = (const int*)d_in[1];
    const int*   relation_ids      = (const int*)d_in[2];
    const int*   meta_relation_ids = (const int*)d_in[3];
    const int*   tail_ids          = (const int*)d_in[4];
    const int*   meta_tail_ids     = (const int*)d_in[5];
    const float* entity_emb        = (const float*)d_in[6];
    const float* relation_emb      = (const float*)d_in[7];
    const float* meta_entity_emb   = (const float*)d_in[8];
    const float* meta_relation_emb = (const float*)d_in[9];

    const int B = in_sizes[0];
    float* partials = (float*)d_ws;   // NBLOCKS floats (4 KB)

    transd_partial_kernel<<<NBLOCKS, BLOCK_THREADS, 0, stream>>>(
        head_ids, meta_head_ids, relation_ids, meta_relation_ids,
        tail_ids, meta_tail_ids,
        entity_emb, relation_emb, meta_entity_emb, meta_relation_emb,
        partials, B);

    final_reduce_kernel<<<1, BLOCK_THREADS, 0, stream>>>(
        partials, (float*)d_out, NBLOCKS);
}